// CRF_2654289789024
// MI455X (gfx1250) — compile-verified
//
#include <hip/hip_runtime.h>
#include <hip/hip_bf16.h>

// Problem constants (match reference)
#define BB 32
#define TT 2048
#define DD 512
#define UU 32
#define RR (BB * TT)   // 65536 rows of the flattened (b,t) GEMM

typedef __attribute__((ext_vector_type(2))) float v2f;
typedef __attribute__((ext_vector_type(8))) float v8f;

// ---------------------------------------------------------------------------
// Kernel 1: e[b,t,u] = sum_d X[b,t,d]*K[d,u] + bias[u] (+boundaries)
// fp32 WMMA 16x16x4. One workgroup = 128 threads = 4 waves = 64 rows.
// LDS staging per 128-wide K chunk:
//   sX : 64 rows x 128 floats (row-major)                     -> 32 KB
//   sKp: K-pair interleaved: sKp[p][c][2] = {K[2p][c],K[2p+1][c]} -> 16 KB
// The pair-interleave makes every WMMA B-fragment a single aligned 8-byte
// LDS load into an even VGPR pair (no repacking movs in the hot loop).
// ---------------------------------------------------------------------------
__global__ __launch_bounds__(128) void crf_gemm_e(
    const float* __restrict__ X, const float* __restrict__ Kmat,
    const float* __restrict__ bias, const float* __restrict__ lb,
    const float* __restrict__ rb, float* __restrict__ eOut)
{
    __shared__ __align__(16) float sX[64 * 128];    // 32 KB
    __shared__ __align__(16) float sKp[64 * 64];    // 16 KB (64 pairs x 32 cols x 2)

    const int tid     = threadIdx.x;       // 0..127
    const int rowBase = blockIdx.x * 64;   // 1024 workgroups cover 65536 rows
    const int wave    = tid >> 5;          // 0..3
    const int lane    = tid & 31;
    const int halfsel = lane >> 4;         // 0: lanes 0-15, 1: lanes 16-31
    const int lrow    = lane & 15;
    const int myRow   = wave * 16 + lrow;  // A-row (within 64-row tile) for this lane

    v8f acc0 = {};   // u = 0..15
    v8f acc1 = {};   // u = 16..31

    for (int kc = 0; kc < DD; kc += 128) {
        __syncthreads();   // WAR vs previous chunk's compute
        // Stage X chunk: 64 rows x 128 floats (float4-coalesced within rows)
        for (int i = tid; i < 64 * 32; i += 128) {
            const int r  = i >> 5;
            const int c4 = i & 31;
            ((float4*)sX)[r * 32 + c4] =
                *(const float4*)(X + (size_t)(rowBase + r) * DD + kc + c4 * 4);
        }
        // Stage kernel chunk, K-pair interleaved: 64 row-pairs x 32 cols.
        // Lanes sweep cols -> coalesced 128B global reads per row.
        for (int i = tid; i < 64 * 32; i += 128) {
            const int p = i >> 5;          // row pair 0..63
            const int c = i & 31;          // col 0..31
            const float v0 = Kmat[(size_t)(kc + p * 2) * UU + c];
            const float v1 = Kmat[(size_t)(kc + p * 2 + 1) * UU + c];
            v2f pk; pk.x = v0; pk.y = v1;
            *(v2f*)(sKp + p * 64 + c * 2) = pk;
        }
        // Speculative prefetch of next X chunk (global_prefetch_b8)
        if (kc + 128 < DD) {
            __builtin_prefetch(X + (size_t)(rowBase + (tid >> 1)) * DD + kc + 128, 0, 0);
        }
        __syncthreads();

        const float* arow = sX + myRow * 128 + halfsel * 2;
        #pragma unroll 8
        for (int k = 0; k < 128; k += 4) {
            // A fragment: lane half 0 holds K={k,k+1}; half 1 holds K={k+2,k+3}
            const v2f a = *(const v2f*)(arow + k);
            // B fragments: rows {k+2h, k+1+2h} = pair (k/2 + h); one b64 each.
            const float* kb = sKp + (k / 2 + halfsel) * 64 + lrow * 2;
            const v2f b0 = *(const v2f*)(kb);        // cols lrow
            const v2f b1 = *(const v2f*)(kb + 32);   // cols 16+lrow
            acc0 = __builtin_amdgcn_wmma_f32_16x16x4_f32(
                false, a, false, b0, (short)0, acc0, false, false);
            acc1 = __builtin_amdgcn_wmma_f32_16x16x4_f32(
                false, a, false, b1, (short)0, acc1, false, false);
        }
    }

    // Epilogue: bias + left/right boundary, store e.
    const float bias0 = bias[lrow],      bias1 = bias[16 + lrow];
    const float lb0   = lb[lrow],        lb1   = lb[16 + lrow];
    const float rb0   = rb[lrow],        rb1   = rb[16 + lrow];
    #pragma unroll
    for (int i = 0; i < 8; ++i) {
        const int row = rowBase + wave * 16 + i + 8 * halfsel;  // C layout: M = i + 8*halfsel
        const int t   = row & (TT - 1);
        float add0 = bias0, add1 = bias1;
        if (t == 0)      { add0 += lb0; add1 += lb1; }
        if (t == TT - 1) { add0 += rb0; add1 += rb1; }
        eOut[(size_t)row * UU + lrow]      = acc0[i] + add0;
        eOut[(size_t)row * UU + 16 + lrow] = acc1[i] + add1;
    }
}

// ---------------------------------------------------------------------------
// Kernel 2: alpha/beta chain recursion. One wave per (batch, direction).
// lane u: new[u] = log sum_v exp(prev[v] - e_t[v] - chain[v][u])
// chain column kept in 32 registers; cross-lane vector via LDS broadcast;
// two-pass (max, then sum-of-exp) logsumexp for stability + short dep chains.
// ---------------------------------------------------------------------------
__global__ __launch_bounds__(32) void crf_recursion(
    const float* __restrict__ e, const float* __restrict__ chain,
    float* __restrict__ alpha, float* __restrict__ beta)
{
    const int b   = blockIdx.x >> 1;
    const int dir = blockIdx.x & 1;       // 0 = forward (alpha), 1 = backward (beta)
    const int u   = threadIdx.x;

    __shared__ float lm[UU];

    float ck[UU];
    #pragma unroll
    for (int v = 0; v < UU; ++v) ck[v] = chain[v * UU + u];   // chain[v][u]

    const float* eb = e + (size_t)b * TT * UU;
    float*       ob = (dir ? beta : alpha) + (size_t)b * TT * UU;

    float prev = 0.0f;
    for (int i = 0; i < TT; ++i) {
        const int t = dir ? (TT - 1 - i) : i;
        const float et = eb[(size_t)t * UU + u];
        lm[u] = prev - et;
        __syncthreads();

        float a[UU];
        float mx = -3.402823466e38f;
        #pragma unroll
        for (int v = 0; v < UU; ++v) {
            a[v] = lm[v] - ck[v];          // uniform-address LDS read -> broadcast
            mx = fmaxf(mx, a[v]);
        }
        float s = 0.0f;
        #pragma unroll
        for (int v = 0; v < UU; ++v) s += __expf(a[v] - mx);

        prev = __logf(s) + mx;
        ob[(size_t)t * UU + u] = prev;
        __syncthreads();                   // protect lm before next overwrite
    }
}

// ---------------------------------------------------------------------------
// Kernel 3: margin = -(alpha[t-1] + e[t] + beta[t+1]); softmax over u.
// One 32-lane wave per (b,t) row; shfl_xor reductions.
// ---------------------------------------------------------------------------
__global__ __launch_bounds__(256) void crf_softmax(
    const float* __restrict__ e, const float* __restrict__ alpha,
    const float* __restrict__ beta, float* __restrict__ out)
{
    const int gid = blockIdx.x * 256 + threadIdx.x;
    const int row = gid >> 5;          // b*T + t
    const int u   = gid & 31;
    const int t   = row & (TT - 1);

    const float asr = (t == 0)      ? 0.0f : alpha[(size_t)(row - 1) * UU + u];
    const float bsl = (t == TT - 1) ? 0.0f : beta[(size_t)(row + 1) * UU + u];
    const float margin = -(asr + e[(size_t)row * UU + u] + bsl);

    float mx = margin;
    #pragma unroll
    for (int off = 16; off; off >>= 1) mx = fmaxf(mx, __shfl_xor(mx, off, 32));
    const float ex = __expf(margin - mx);
    float s = ex;
    #pragma unroll
    for (int off = 16; off; off >>= 1) s += __shfl_xor(s, off, 32);

    out[gid] = ex / s;
}

// ---------------------------------------------------------------------------
extern "C" void kernel_launch(void* const* d_in, const int* in_sizes, int n_in,
                              void* d_out, int out_size, void* d_ws, size_t ws_size,
                              hipStream_t stream) {
    const float* X     = (const float*)d_in[0];   // (B,T,D)
    const float* kern  = (const float*)d_in[1];   // (D,U)
    const float* chain = (const float*)d_in[2];   // (U,U)
    const float* bias  = (const float*)d_in[3];   // (U,)
    const float* lb    = (const float*)d_in[4];   // (U,)
    const float* rb    = (const float*)d_in[5];   // (U,)

    const size_t nE = (size_t)BB * TT * UU;       // 2,097,152 floats (8 MB)
    float* e     = (float*)d_ws;                  // ws usage: 24 MB total
    float* alpha = e + nE;
    float* beta  = alpha + nE;
    float* out   = (float*)d_out;

    crf_gemm_e<<<RR / 64, 128, 0, stream>>>(X, kern, bias, lb, rb, e);
    crf_recursion<<<BB * 2, 32, 0, stream>>>(e, chain, alpha, beta);
    crf_softmax<<<(RR * UU) / 256, 256, 0, stream>>>(e, alpha, beta, out);
}